// GaborRendererCUDAOptimized_76287209112275
// MI455X (gfx1250) — compile-verified
//
#include <hip/hip_runtime.h>

// Gabor atom renderer for MI455X (gfx1250, wave32).
// Gather-by-tile: one workgroup owns a 512-sample output tile in LDS.
// Atoms are chunk-staged via CDNA5 async global->LDS copies, compacted per
// tile, then rendered wave-parallel (one atom per wave at a time, lanes
// stride samples) with ds_add_f32 LDS atomics. Tile stored exactly once.

#define SR      24000.0f
#define INV_SR  (1.0f / 24000.0f)
#define TWO_PI  6.28318530717958647692f
#define TILE    512
#define CHUNK   2048
#define BLOCK   256
#define NWAVES  (BLOCK / 32)
#define MAXW    4096

__global__ __launch_bounds__(BLOCK) void gabor_tile_kernel(
    const float* __restrict__ amp,   const float* __restrict__ tau,
    const float* __restrict__ omega, const float* __restrict__ sigma,
    const float* __restrict__ phi,   const float* __restrict__ gamma,
    float* __restrict__ out, int num_samples, int num_atoms)
{
    __shared__ float s_acc[TILE];
    __shared__ float s_tau[CHUNK];
    __shared__ float s_sig[CHUNK];
    __shared__ int   s_list[CHUNK];
    __shared__ int   s_cnt;

    const int tid     = threadIdx.x;
    const int wave    = tid >> 5;   // wave32
    const int lane    = tid & 31;
    const int tile_lo = blockIdx.x * TILE;
    if (tile_lo >= num_samples) return;
    const int tile_hi = min(tile_lo + TILE, num_samples) - 1;

    for (int i = tid; i < TILE; i += BLOCK) s_acc[i] = 0.0f;

    for (int base = 0; base < num_atoms; base += CHUNK) {
        const int n = min(CHUNK, num_atoms - base);

        // ---- CDNA5 async global->LDS staging of tau/sigma chunk ----
        for (int i = tid; i < n; i += BLOCK) {
            unsigned     lt = (unsigned)(unsigned long long)(const void*)&s_tau[i];
            unsigned     ls = (unsigned)(unsigned long long)(const void*)&s_sig[i];
            const float* gt = tau   + base + i;
            const float* gs = sigma + base + i;
            asm volatile("global_load_async_to_lds_b32 %0, %1, off"
                         :: "v"(lt), "v"(gt) : "memory");
            asm volatile("global_load_async_to_lds_b32 %0, %1, off"
                         :: "v"(ls), "v"(gs) : "memory");
        }
        asm volatile("s_wait_asynccnt 0" ::: "memory");
        if (tid == 0) s_cnt = 0;
        __syncthreads();

        // ---- compact atoms whose window intersects this tile ----
        for (int i = tid; i < n; i += BLOCK) {
            float ts = s_tau[i] * SR;
            float hw = fminf(fmaxf(s_sig[i] * SR * 4.0f, 1.0f), 2048.0f);
            int start = max((int)(ts - hw), 0);              // trunc-toward-zero == ref
            int end   = min((int)(ts + hw), num_samples - 1);
            end = min(end, start + (MAXW - 1));              // local_idx < 4096 cap
            if (start <= tile_hi && end >= tile_lo) {
                int pos = atomicAdd(&s_cnt, 1);              // ds_add_rtn_u32
                s_list[pos] = i;
            }
        }
        __syncthreads();

        // ---- render: one atom per wave, lanes stride samples ----
        const int cnt = s_cnt;
        for (int j = wave; j < cnt; j += NWAVES) {
            // list index j is uniform per wave -> force scalar param loads
            const int   li  = __builtin_amdgcn_readfirstlane(s_list[j]);
            const int   idx = base + li;
            const float tv  = s_tau[li];
            const float sv  = s_sig[li];
            const float av  = amp[idx];
            const float ov  = omega[idx];
            const float pv  = phi[idx];
            const float gv  = gamma[idx];

            float ts = tv * SR;
            float hw = fminf(fmaxf(sv * SR * 4.0f, 1.0f), 2048.0f);
            int start = max((int)(ts - hw), 0);
            int end   = min((int)(ts + hw), num_samples - 1);
            end = min(end, start + (MAXW - 1));

            const int   lo      = max(start, tile_lo);
            const int   hi      = min(end,   tile_hi);
            const float inv_den = 1.0f / (2.0f * sv * sv + 1e-8f);

            for (int g = lo + lane; g <= hi; g += 32) {
                float tc  = (float)g * INV_SR - tv;
                float tc2 = tc * tc;
                float env = __expf(-tc2 * inv_den);              // v_exp_f32
                float ph  = TWO_PI * fmaf(0.5f * gv, tc2, ov * tc) + pv;
                float c   = av * env * cosf(ph);                 // precise reduction
                atomicAdd(&s_acc[g - tile_lo], c);               // ds_add_f32
            }
        }
        __syncthreads();
    }

    // ---- store tile exactly once (no global atomics, no pre-zeroing) ----
    for (int i = tid; i <= tile_hi - tile_lo; i += BLOCK)
        out[tile_lo + i] = s_acc[i];
}

extern "C" void kernel_launch(void* const* d_in, const int* in_sizes, int n_in,
                              void* d_out, int out_size, void* d_ws, size_t ws_size,
                              hipStream_t stream) {
    const float* amp   = (const float*)d_in[0];
    const float* tau   = (const float*)d_in[1];
    const float* omega = (const float*)d_in[2];
    const float* sigma = (const float*)d_in[3];
    const float* phi   = (const float*)d_in[4];
    const float* gamma = (const float*)d_in[5];
    float*       out   = (float*)d_out;

    const int num_atoms   = in_sizes[0];
    const int num_samples = out_size;   // output length == num_samples
    const int grid        = (num_samples + TILE - 1) / TILE;

    hipLaunchKernelGGL(gabor_tile_kernel, dim3(grid), dim3(BLOCK), 0, stream,
                       amp, tau, omega, sigma, phi, gamma, out,
                       num_samples, num_atoms);
}